// GCN_29386166239463
// MI455X (gfx1250) — compile-verified
//
#include <hip/hip_runtime.h>
#include <math.h>

#define NDIM 128
#define HID  16
#define NCLS 40

typedef __attribute__((ext_vector_type(2))) float v2f;
typedef __attribute__((ext_vector_type(8))) float v8f;

// ---------------- init: deg=1 (self loop), zero accumulators ----------------
__global__ void gcn_init(float* deg, float* agg1, float* out, int n) {
  int idx = blockIdx.x * blockDim.x + threadIdx.x;
  if (idx < n * NCLS) out[idx]  = 0.f;
  if (idx < n * HID)  agg1[idx] = 0.f;
  if (idx < n)        deg[idx]  = 1.f;
}

// ---------------- degree count over edge destinations ----------------
__global__ void gcn_deg(const int* __restrict__ dst, float* deg, int e) {
  int idx = blockIdx.x * blockDim.x + threadIdx.x;
  if (idx < e) atomicAdd(&deg[dst[idx]], 1.0f);
}

__global__ void gcn_dinv(const float* __restrict__ deg, float* dinv, int n) {
  int idx = blockIdx.x * blockDim.x + threadIdx.x;
  if (idx < n) dinv[idx] = rsqrtf(deg[idx]);   // deg >= 1 always (self loop)
}

// ---------------- GEMM1: h1[N,16] = x[N,128] @ W1[128,16] via WMMA f32 16x16x4 ----------------
// One wave computes a 16x16 output tile (16 nodes x HID). K swept in steps of 4.
// A layout (16x4 f32): lanes 0-15 hold row m, K = k0,k0+1 ; lanes 16-31 hold row m, K = k0+2,k0+3.
// B layout (4x16 f32): v0 lanes0-15 = row k0 / lanes16-31 = row k0+2; v1 = k0+1 / k0+3.
__global__ void gcn_gemm1(const float* __restrict__ x, const float* __restrict__ W1,
                          float* __restrict__ h1, int n, int ntiles) {
  __shared__ float w1s[NDIM * HID];
  for (int j = threadIdx.x; j < NDIM * HID; j += blockDim.x) w1s[j] = W1[j];
  __syncthreads();

  int wave = blockIdx.x * (blockDim.x >> 5) + (threadIdx.x >> 5);
  if (wave >= ntiles) return;                      // wave-uniform exit, EXEC stays full
  int lane = threadIdx.x & 31;
  int m  = lane & 15;
  int hf = lane >> 4;                              // which K-pair this lane holds
  int row0 = wave * 16;
  int row  = row0 + m; if (row >= n) row = n - 1;  // clamp instead of masking (EXEC must be all 1s)

  const float* xr = x + (size_t)row * NDIM;
  v8f c = {};
  for (int k0 = 0; k0 < NDIM; k0 += 4) {
    int ka = k0 + hf * 2;
    v2f a; a.x = xr[ka];               a.y = xr[ka + 1];
    v2f b; b.x = w1s[ka * HID + m];    b.y = w1s[(ka + 1) * HID + m];
    c = __builtin_amdgcn_wmma_f32_16x16x4_f32(false, a, false, b, (short)0, c, false, false);
  }
  for (int v = 0; v < 8; v++) {
    int r = row0 + v + hf * 8;                     // C/D layout: vgpr v = row v (+8 for hi lanes)
    if (r < n) h1[r * HID + m] = c[v];
  }
}

// ---------------- edge aggregation layer 1: 4 threads/edge, float4 each ----------------
__global__ void gcn_agg1(const int* __restrict__ src, const int* __restrict__ dst,
                         const float* __restrict__ dinv, const float* __restrict__ h1,
                         float* agg1, int e) {
  int idx = blockIdx.x * blockDim.x + threadIdx.x;
  if (idx >= e * 4) return;
  int ed = idx >> 2, q = idx & 3;
  int s = src[ed], d = dst[ed];
  float w = dinv[s] * dinv[d];
  const float4 hv = *reinterpret_cast<const float4*>(h1 + s * HID + q * 4);
  float* ap = agg1 + d * HID + q * 4;
  atomicAdd(ap + 0, hv.x * w);
  atomicAdd(ap + 1, hv.y * w);
  atomicAdd(ap + 2, hv.z * w);
  atomicAdd(ap + 3, hv.w * w);
}

// ---------------- self-loop + bias + ReLU (agg1 -> h1r in place) ----------------
__global__ void gcn_bias_relu(float* agg1, const float* __restrict__ h1,
                              const float* __restrict__ dinv, const float* __restrict__ b1, int n) {
  int idx = blockIdx.x * blockDim.x + threadIdx.x;
  if (idx >= n * HID) return;
  int i = idx >> 4, f = idx & 15;
  float di = dinv[i];
  float v = agg1[idx] + h1[idx] * di * di + b1[f];
  agg1[idx] = v > 0.f ? v : 0.f;
}

// ---------------- GEMM2: h2[N,40] = h1r[N,16] @ W2[16,40]; 3 col-tiles of 16 (padded) ----------------
__global__ void gcn_gemm2(const float* __restrict__ h1r, const float* __restrict__ W2,
                          float* __restrict__ h2, int n, int ntasks) {
  __shared__ float w2s[HID * 48];                   // W2 padded 16x48
  for (int j = threadIdx.x; j < HID * 48; j += blockDim.x) {
    int r = j / 48, cc = j % 48;
    w2s[j] = (cc < NCLS) ? W2[r * NCLS + cc] : 0.f;
  }
  __syncthreads();

  int wave = blockIdx.x * (blockDim.x >> 5) + (threadIdx.x >> 5);
  if (wave >= ntasks) return;
  int rt = wave / 3, ct = wave % 3;
  int lane = threadIdx.x & 31;
  int m  = lane & 15;
  int hf = lane >> 4;
  int row0 = rt * 16, col0 = ct * 16;
  int row  = row0 + m; if (row >= n) row = n - 1;

  const float* hr = h1r + (size_t)row * HID;
  v8f c = {};
  for (int k0 = 0; k0 < HID; k0 += 4) {
    int ka = k0 + hf * 2;
    v2f a; a.x = hr[ka];                     a.y = hr[ka + 1];
    v2f b; b.x = w2s[ka * 48 + col0 + m];    b.y = w2s[(ka + 1) * 48 + col0 + m];
    c = __builtin_amdgcn_wmma_f32_16x16x4_f32(false, a, false, b, (short)0, c, false, false);
  }
  int col = col0 + m;
  if (col < NCLS) {
    for (int v = 0; v < 8; v++) {
      int r = row0 + v + hf * 8;
      if (r < n) h2[r * NCLS + col] = c[v];
    }
  }
}

// ---------------- edge aggregation layer 2: 10 threads/edge, float4 each ----------------
__global__ void gcn_agg2(const int* __restrict__ src, const int* __restrict__ dst,
                         const float* __restrict__ dinv, const float* __restrict__ h2,
                         float* out, int e) {
  int idx = blockIdx.x * blockDim.x + threadIdx.x;
  if (idx >= e * 10) return;
  int ed = idx / 10, q = idx - ed * 10;
  int s = src[ed], d = dst[ed];
  float w = dinv[s] * dinv[d];
  const float4 hv = *reinterpret_cast<const float4*>(h2 + s * NCLS + q * 4);
  float* op = out + d * NCLS + q * 4;
  atomicAdd(op + 0, hv.x * w);
  atomicAdd(op + 1, hv.y * w);
  atomicAdd(op + 2, hv.z * w);
  atomicAdd(op + 3, hv.w * w);
}

// ---------------- fused self-loop + bias + log_softmax (in place on out) ----------------
__global__ void gcn_logsoftmax(float* out, const float* __restrict__ h2,
                               const float* __restrict__ dinv, const float* __restrict__ b2, int n) {
  int i = blockIdx.x * blockDim.x + threadIdx.x;
  if (i >= n) return;
  float di = dinv[i];
  float dii = di * di;
  float v[NCLS];
  float mx = -INFINITY;
  const float* orow = out + (size_t)i * NCLS;
  const float* hrow = h2 + (size_t)i * NCLS;
  for (int f = 0; f < NCLS; f++) {
    float t = orow[f] + hrow[f] * dii + b2[f];
    v[f] = t;
    mx = fmaxf(mx, t);
  }
  float s = 0.f;
  for (int f = 0; f < NCLS; f++) s += __expf(v[f] - mx);
  float lse = __logf(s) + mx;
  float* wrow = out + (size_t)i * NCLS;
  for (int f = 0; f < NCLS; f++) wrow[f] = v[f] - lse;
}

extern "C" void kernel_launch(void* const* d_in, const int* in_sizes, int n_in,
                              void* d_out, int out_size, void* d_ws, size_t ws_size,
                              hipStream_t stream) {
  const float* x   = (const float*)d_in[0];
  const int*   ei  = (const int*)  d_in[1];     // [2,E] row-major: src then dst
  const float* W1  = (const float*)d_in[2];
  const float* b1  = (const float*)d_in[3];
  const float* W2  = (const float*)d_in[4];
  const float* b2  = (const float*)d_in[5];
  float* out = (float*)d_out;

  const int n = in_sizes[0] / NDIM;
  const int e = in_sizes[1] / 2;
  const int* src = ei;
  const int* dst = ei + e;

  // workspace layout (floats): deg[n] | dinv[n] | h1[n*16] | agg1[n*16] | h2[n*40]
  float* deg  = (float*)d_ws;
  float* dinv = deg  + n;
  float* h1   = dinv + n;
  float* agg1 = h1   + (size_t)n * HID;
  float* h2   = agg1 + (size_t)n * HID;

  const int B = 256;
  int ntiles = (n + 15) / 16;
  int ntasks = ntiles * 3;

  gcn_init     <<<(n * NCLS + B - 1) / B, B, 0, stream>>>(deg, agg1, out, n);
  gcn_deg      <<<(e + B - 1) / B,         B, 0, stream>>>(dst, deg, e);
  gcn_dinv     <<<(n + B - 1) / B,         B, 0, stream>>>(deg, dinv, n);
  gcn_gemm1    <<<(ntiles + 7) / 8,        B, 0, stream>>>(x, W1, h1, n, ntiles);
  gcn_agg1     <<<(e * 4 + B - 1) / B,     B, 0, stream>>>(src, dst, dinv, h1, agg1, e);
  gcn_bias_relu<<<(n * HID + B - 1) / B,   B, 0, stream>>>(agg1, h1, dinv, b1, n);
  gcn_gemm2    <<<(ntasks + 7) / 8,        B, 0, stream>>>(agg1, W2, h2, n, ntasks);
  gcn_agg2     <<<(e * 10 + B - 1) / B,    B, 0, stream>>>(src, dst, dinv, h2, out, e);
  gcn_logsoftmax<<<(n + B - 1) / B,        B, 0, stream>>>(out, h2, dinv, b2, n);
}